// RF_tensor_cat_2559800508493
// MI455X (gfx1250) — compile-verified
//
#include <hip/hip_runtime.h>
#include <math.h>

// Rank-4 outer-product factorization of the reference PDE expression,
// evaluated with V_WMMA_F32_16X16X4_F32 (the native K=4 f32 WMMA on gfx1250).
//
//   out[i][j] = sum_k A[i][k] * B[j][k]   (exact algebraic decomposition)
//
// Store-bandwidth bound: 268 MB of output writes -> ~11.5 us floor at 23.3 TB/s.

typedef __attribute__((ext_vector_type(2))) float v2f;
typedef __attribute__((ext_vector_type(8))) float v8f;

// ---------------------------------------------------------------------------
// Kernel 1: build the rank-4 factor tables (A from qx1, B from qx2).
//   u  = exp(-(x^2 + 2y^2 + 1)) = ex * ey,  ex = exp(-(x^2+1)), ey = exp(-2y^2)
//   out = -4x^2(1+2y^2) u^2 - 16y^2(1+2y^2) u^2 + (7 + 28y^2 + x^2) u
// ---------------------------------------------------------------------------
__global__ __launch_bounds__(256) void rf_build_factors(
    const float* __restrict__ qx1, const float* __restrict__ qx2,
    float* __restrict__ Am, float* __restrict__ Bm, int n) {
  int i = blockIdx.x * blockDim.x + threadIdx.x;
  if (i >= n) return;

  float x   = qx1[i];
  float x2  = x * x;
  float ex  = expf(-(x2 + 1.0f));
  float ex2 = ex * ex;
  Am[i * 4 + 0] = -4.0f * x2 * ex2;   // pairs with (1+2y^2) ey^2
  Am[i * 4 + 1] = -ex2;               // pairs with 16 y^2 (1+2y^2) ey^2
  Am[i * 4 + 2] = ex;                 // pairs with (7+28y^2) ey
  Am[i * 4 + 3] = x2 * ex;            // pairs with ey

  float y   = qx2[i];
  float y2  = y * y;
  float ey  = expf(-2.0f * y2);
  float ey2 = ey * ey;
  float t   = 1.0f + 2.0f * y2;
  Bm[i * 4 + 0] = t * ey2;
  Bm[i * 4 + 1] = 16.0f * y2 * t * ey2;
  Bm[i * 4 + 2] = (7.0f + 28.0f * y2) * ey;
  Bm[i * 4 + 3] = ey;
}

// ---------------------------------------------------------------------------
// Kernel 2 (specialized, N=8192): 4 column-tiles per wave, one shared A frag.
//
// A-fragment (16x4 f32, 2 VGPRs): lane<16 -> row (r0+lane), K=0..1;
// lane>=16 -> row (r0+lane-16), K=2..3.  Tables stored [index][4] make each
// lane's fragment a single 8B-aligned global_load_b64.  B 4x16 mirrors the
// striping over columns; K-pairing is consistent, so D = A.B^T.
//
// Compile-time stride lets all 32 stores share one 64-bit address with
// immediate offsets (v*8192*4 + t*64 bytes < 2^23).
// ---------------------------------------------------------------------------
__global__ __launch_bounds__(256) void rf_rank4_wmma_tile4(
    const float* __restrict__ Am, const float* __restrict__ Bm,
    float* __restrict__ out) {
  constexpr int N = 8192;
  const int lane = threadIdx.x & 31;
  const int wave = threadIdx.x >> 5;       // 8 waves per block
  const int l15  = lane & 15;
  const int half = lane >> 4;              // 0: K=0..1, 1: K=2..3

  const int r0 = blockIdx.y << 4;                       // tile row base
  const int c0 = (blockIdx.x * 8 + wave) << 6;          // 4 tiles * 16 cols

  // One A fragment shared by all 4 WMMAs.
  v2f a = *(const v2f*)(Am + (size_t)(r0 + l15) * 4 + half * 2);

  v2f b[4];
#pragma unroll
  for (int t = 0; t < 4; ++t)
    b[t] = *(const v2f*)(Bm + (size_t)(c0 + t * 16 + l15) * 4 + half * 2);

  v8f acc[4];
#pragma unroll
  for (int t = 0; t < 4; ++t) {
    v8f c = {};
    // (neg_a, A, neg_b, B, c_mod, C, reuse_a, reuse_b)
    acc[t] = __builtin_amdgcn_wmma_f32_16x16x4_f32(
        false, a, false, b[t], (short)0, c, false, false);
  }

  // D layout: VGPR v -> row v (lanes 0-15) / row v+8 (lanes 16-31), col = l15.
  float* base = out + (size_t)(r0 + half * 8) * N + (size_t)(c0 + l15);
#pragma unroll
  for (int t = 0; t < 4; ++t) {
#pragma unroll
    for (int v = 0; v < 8; ++v) {
      base[v * N + t * 16] = acc[t][v];   // constant byte offsets -> saddr-free
    }
  }
}

// ---------------------------------------------------------------------------
// Generic fallback (runtime stride), one tile per wave.
// ---------------------------------------------------------------------------
__global__ __launch_bounds__(256) void rf_rank4_wmma_tile(
    const float* __restrict__ Am, const float* __restrict__ Bm,
    float* __restrict__ out, int n) {
  const int lane = threadIdx.x & 31;
  const int wave = threadIdx.x >> 5;
  const int l15  = lane & 15;
  const int half = lane >> 4;

  const int r0 = blockIdx.y << 4;
  const int c0 = (blockIdx.x * 8 + wave) << 4;

  v2f a = *(const v2f*)(Am + (size_t)(r0 + l15) * 4 + half * 2);
  v2f b = *(const v2f*)(Bm + (size_t)(c0 + l15) * 4 + half * 2);

  v8f c = {};
  c = __builtin_amdgcn_wmma_f32_16x16x4_f32(
      false, a, false, b, (short)0, c, false, false);

  const size_t stride = (size_t)n;
  size_t base = (size_t)(r0 + half * 8) * stride + (size_t)(c0 + l15);
#pragma unroll
  for (int v = 0; v < 8; ++v) {
    out[base] = c[v];
    base += stride;
  }
}

extern "C" void kernel_launch(void* const* d_in, const int* in_sizes, int n_in,
                              void* d_out, int out_size, void* d_ws, size_t ws_size,
                              hipStream_t stream) {
  const float* qx1 = (const float*)d_in[0];
  const float* qx2 = (const float*)d_in[1];
  // d_in[2] (param_a) is unused by the reference expression.
  float* out = (float*)d_out;

  const int N = in_sizes[0];               // 8192

  float* Am = (float*)d_ws;                // N*4 floats
  float* Bm = Am + (size_t)N * 4;          // N*4 floats  (total 256 KB)

  rf_build_factors<<<(N + 255) / 256, 256, 0, stream>>>(qx1, qx2, Am, Bm, N);

  if (N == 8192) {
    // 512x512 tiles of 16x16; each wave does 4 col-tiles; 8 waves per block.
    dim3 grid(8192 / 16 / (4 * 8), 8192 / 16);   // (16, 512)
    rf_rank4_wmma_tile4<<<grid, 256, 0, stream>>>(Am, Bm, out);
  } else {
    dim3 grid(N / 16 / 8, N / 16);
    rf_rank4_wmma_tile<<<grid, 256, 0, stream>>>(Am, Bm, out, N);
  }
}